// GraphConvLayer_1726576853372
// MI455X (gfx1250) — compile-verified
//
#include <hip/hip_runtime.h>
#include <hip/hip_bf16.h>
#include <math.h>

// Problem constants (from reference): B=8, S=128, E=50, D=300, DE=50
#define BB 8
#define SS 128
#define EE 50
#define DD 300
#define DEE 50
#define DP 320          // DD padded to a multiple of 32 (K) and 16 (N)

typedef float v2f __attribute__((ext_vector_type(2)));
typedef float v8f __attribute__((ext_vector_type(8)));

// ---------------------------------------------------------------------------
// Kernel 1: A[b,i,j] = (1/E) * sum_e softmax[b,i,j,e] + (i==j)
// One block per (b,i) row; stage the [128 x 50] slab through LDS so the
// global reads are perfectly coalesced (E is innermost in memory).
// ---------------------------------------------------------------------------
__global__ void k_reduce_wps(const float* __restrict__ wps, float* __restrict__ Amat) {
    __shared__ float tile[SS * EE];            // 25.6 KB of the 320 KB WGP LDS
    int bi = blockIdx.x;                       // b*128 + i
    int i  = bi & (SS - 1);
    const float* src = wps + (size_t)bi * (SS * EE);
    for (int idx = threadIdx.x; idx < SS * EE; idx += 256)
        tile[idx] = src[idx];
    __syncthreads();
    if (threadIdx.x < SS) {
        int j = threadIdx.x;
        float s = 0.f;
        #pragma unroll
        for (int e = 0; e < EE; ++e) s += tile[j * EE + e];
        Amat[(size_t)bi * SS + j] = s * (1.0f / (float)EE) + (j == i ? 1.0f : 0.0f);
    }
}

// ---------------------------------------------------------------------------
// Generic strided-batched fp32 GEMM using V_WMMA_F32_16X16X4_F32.
// HARD REQUIREMENTS (guaranteed by padded operands): M%64==0, N%16==0,
// K%32==0, all row strides %4==0, all base pointers 16B aligned.
// -> staging and epilogue are completely branch-free and mask-free.
// Block = 128 threads = 4 waves -> one 64x16 C tile. K tiled by 32.
// Staging: registers first (all global loads in flight), then LDS.
//   A: 4 x float4 per thread (8 lanes cover a 128B row segment, coalesced)
//   B: 1 x float4 per thread
// Fragment layouts per ISA 7.12.2:
//   A (16x4):  lane L holds row M=L&15, K = 2*(L>>4)+v   (v = VGPR 0/1)
//   B (4x16):  lane L holds col N=L&15, K = 2*(L>>4)+v
//   C (16x16): VGPR r -> M = r + 8*(L>>4), N = L&15
// ---------------------------------------------------------------------------
#define KT 32
#define ASTR 36   // LDS row stride for A tile: 16B-aligned rows, 36r%64=4*(9r%16)
                  // distinct banks for r=0..15; half-waves in disjoint mod-4 classes.
__global__ void k_gemm_wmma(const float* __restrict__ A, int lda, long sA,
                            const float* __restrict__ Bm, int ldb, long sB,
                            float* __restrict__ C, int ldc, long sC,
                            int M, int N, int K, const float* __restrict__ bias) {
    __shared__ float Alds[64 * ASTR];          // 9.2 KB
    __shared__ float Blds[KT * 16];            // 2 KB

    int tid  = threadIdx.x;
    int lane = tid & 31;
    int wave = tid >> 5;
    int mBase = blockIdx.x * 64;
    int nBase = blockIdx.y * 16;
    const float* Ab = A  + (size_t)blockIdx.z * sA;
    const float* Bb = Bm + (size_t)blockIdx.z * sB;
    float*       Cb = C  + (size_t)blockIdx.z * sC;

    int mloc  = wave * 16 + (lane & 15);
    int cloc  = lane & 15;
    int khalf = 2 * (lane >> 4);
    int ncol  = nBase + cloc;

    // per-thread staging coordinates (constant across K tiles)
    int ar[4], ak[4];
    #pragma unroll
    for (int it = 0; it < 4; ++it) {
        int idx = it * 128 + tid;              // 512 float4s = 64 rows x 8
        ar[it] = idx >> 3;
        ak[it] = (idx & 7) * 4;
    }
    int bkk = tid >> 2;                        // 0..31
    int bc4 = (tid & 3) * 4;                   // 0,4,8,12

    v8f acc = {};
    for (int k0 = 0; k0 < K; k0 += KT) {
        // ---- all global loads first (no masks, no waits in between) ----
        float4 av[4];
        #pragma unroll
        for (int it = 0; it < 4; ++it)
            av[it] = *(const float4*)&Ab[(size_t)(mBase + ar[it]) * lda + k0 + ak[it]];
        float4 bv4 = *(const float4*)&Bb[(size_t)(k0 + bkk) * ldb + nBase + bc4];
        // ---- then LDS stores ----
        #pragma unroll
        for (int it = 0; it < 4; ++it)
            *(float4*)&Alds[ar[it] * ASTR + ak[it]] = av[it];
        *(float4*)&Blds[bkk * 16 + bc4] = bv4;
        __syncthreads();
        // ---- 8 back-to-back WMMAs fed purely from LDS ----
        #pragma unroll
        for (int kk0 = 0; kk0 < KT; kk0 += 4) {
            int kk = kk0 + khalf;
            v2f a, b;
            a.x = Alds[mloc * ASTR + kk];
            a.y = Alds[mloc * ASTR + kk + 1];
            b.x = Blds[kk * 16 + cloc];
            b.y = Blds[(kk + 1) * 16 + cloc];
            acc = __builtin_amdgcn_wmma_f32_16x16x4_f32(false, a, false, b,
                                                        (short)0, acc, false, false);
        }
        __syncthreads();
    }

    float bv = 0.f;
    if (bias != nullptr) bv = bias[ncol];      // bias is padded to N
    #pragma unroll
    for (int r = 0; r < 8; ++r) {
        int m = mBase + wave * 16 + r + 8 * (lane >> 4);
        Cb[(size_t)m * ldc + ncol] = acc[r] + bv;
    }
}

// ---------------------------------------------------------------------------
// LayerNorm (unbiased std, torch-style (std+eps)) + ReLU.
// One 128-thread block per row; reads padded H (ld=DP), writes packed node.
// ---------------------------------------------------------------------------
__global__ void k_layernorm_relu(const float* __restrict__ H,
                                 const float* __restrict__ ga,
                                 const float* __restrict__ gb,
                                 float* __restrict__ node) {
    __shared__ float red[128];
    int row = blockIdx.x;
    const float* h = H + (size_t)row * DP;

    float s = 0.f;
    for (int c = threadIdx.x; c < DD; c += 128) s += h[c];
    red[threadIdx.x] = s; __syncthreads();
    for (int off = 64; off > 0; off >>= 1) {
        if (threadIdx.x < off) red[threadIdx.x] += red[threadIdx.x + off];
        __syncthreads();
    }
    float mean = red[0] * (1.0f / (float)DD);
    __syncthreads();

    float s2 = 0.f;
    for (int c = threadIdx.x; c < DD; c += 128) {
        float d = h[c] - mean; s2 += d * d;
    }
    red[threadIdx.x] = s2; __syncthreads();
    for (int off = 64; off > 0; off >>= 1) {
        if (threadIdx.x < off) red[threadIdx.x] += red[threadIdx.x + off];
        __syncthreads();
    }
    float stdv = sqrtf(red[0] * (1.0f / (float)(DD - 1)));
    float inv = 1.0f / (stdv + 1e-6f);
    for (int c = threadIdx.x; c < DD; c += 128) {
        float v = ga[c] * (h[c] - mean) * inv + gb[c];
        node[(size_t)row * DD + c] = fmaxf(v, 0.f);
    }
}

// ---------------------------------------------------------------------------
// One-shot pad/pack kernels (tiny, not perf-relevant).
// ---------------------------------------------------------------------------
// gcn_inputs [1024 x 300] -> xpad [1024 x 320] (zero pad cols)
__global__ void k_pad_x_in(const float* __restrict__ x, float* __restrict__ xpad) {
    int row = blockIdx.x;
    for (int c = threadIdx.x; c < DP; c += 256) {
        float v = (c < DD) ? x[(size_t)row * DD + c] : 0.f;
        xpad[(size_t)row * DP + c] = v;
    }
}

// W_w [300x300] -> W_wpad [320x320]; W_b [300] -> W_bpad [320]
__global__ void k_pad_w(const float* __restrict__ Ww, const float* __restrict__ Wb,
                        float* __restrict__ Wwpad, float* __restrict__ Wbpad) {
    int idx = blockIdx.x * 256 + threadIdx.x;
    if (idx < DP * DP) {
        int r = idx / DP, c = idx - r * DP;
        Wwpad[idx] = (r < DD && c < DD) ? Ww[r * DD + c] : 0.f;
    }
    if (idx < DP) Wbpad[idx] = (idx < DD) ? Wb[idx] : 0.f;
}

// X[row=b*128+s] = [ diag(b,s,0:50) | node(b,s,0:300) | 0 0 ]  (352 cols)
__global__ void k_pack_x(const float* __restrict__ wadj,
                         const float* __restrict__ node,
                         float* __restrict__ X) {
    int row = blockIdx.x;                      // b*128 + s
    int b = row >> 7, s = row & (SS - 1);
    const float* dsrc = wadj + (((size_t)(b * SS + s) * SS + s) * EE);
    for (int c = threadIdx.x; c < 352; c += 128) {
        float v;
        if (c < EE)            v = dsrc[c];
        else if (c < EE + DD)  v = node[(size_t)row * DD + (c - EE)];
        else                   v = 0.f;
        X[(size_t)row * 352 + c] = v;
    }
}

// Wcat[352 x 112]: rows 0..49 -> [W2 | W3 | 0]; rows 50..349 -> [W4 | W5 | 0].
// (refine_W row blocks: 0:50=W1(adj), 50:100=W2(edge_i), 100:150=W3(edge_j),
//  150:450=W4(node1=node[j]), 450:750=W5(node2=node[i]).)
__global__ void k_pack_wcat(const float* __restrict__ rW, float* __restrict__ Wcat) {
    int idx = blockIdx.x * 256 + threadIdx.x;
    if (idx >= 352 * 112) return;
    int r = idx / 112, c = idx - r * 112;
    float v = 0.f;
    if (c < 100) {
        if (r < 50) {
            v = (c < 50) ? rW[(50 + r) * DEE + c] : rW[(100 + r) * DEE + (c - 50)];
        } else if (r < 350) {
            int k = r - 50;
            v = (c < 50) ? rW[(150 + k) * DEE + c] : rW[(450 + k) * DEE + (c - 50)];
        }
    }
    Wcat[idx] = v;
}

// W1 (refine_W rows 0..49) zero-padded to a contiguous [52 x 64] image
__global__ void k_pack_w1(const float* __restrict__ rW, float* __restrict__ W1pad) {
    int idx = blockIdx.x * 256 + threadIdx.x;
    if (idx < 52 * 64) {
        int k = idx >> 6, c = idx & 63;
        W1pad[idx] = (k < EE && c < DEE) ? rW[k * DEE + c] : 0.f;
    }
}

// ---------------------------------------------------------------------------
// Dominant, memory-bound kernel: per edge (b,i,j):
//   edge_out[b,i,j,c] = wadj[b,i,j,:] @ W1[:,c] + Pj[b,j,c] + Pi[b,i,c] + rb[c]
// 256 threads/block = 8 waves; block covers 32 consecutive flat rows.
// A rows staged via float2 register batch (contiguous coalesced stream);
// W1pad staged with 13 exact unconditional b32 loads; 13 B fragments
// preloaded to registers -> the unrolled WMMA loop is pure ds_load + v_wmma.
// Wave w: mtile = w>>2 (16 rows), ntile = w&3 (cols 16*nt.., masked to N=50).
// ---------------------------------------------------------------------------
__global__ void k_edge_out(const float* __restrict__ wadj,
                           const float* __restrict__ W1pad, // [52 x 64] contiguous
                           const float* __restrict__ P,     // [1024 x 112]: [:,0:50]=Pj, [:,50:100]=Pi
                           const float* __restrict__ rb,
                           float* __restrict__ eout) {
    __shared__ float tile[32 * 52];            // 6.5 KB  (A rows, stride 52)
    __shared__ float wlds[52 * 64];            // 13.3 KB (W1 padded image)
    int tid  = threadIdx.x;
    int lane = tid & 31;
    int wave = tid >> 5;
    long rowbase = (long)blockIdx.x * 32;

    // ---- all global loads for staging first, then LDS stores ----
    // A: 32 rows x 50 floats = 800 float2 (pairs never straddle rows: 50 even)
    const float2* src2 = (const float2*)(wadj + rowbase * EE);
    float2 areg[3];
    #pragma unroll
    for (int it = 0; it < 3; ++it) areg[it] = src2[it * 256 + tid];
    int ttid = (tid < 32) ? tid : 0;           // clamped (safe) tail load
    float2 atail = src2[768 + ttid];
    // W1: 3328 floats = 13 exact iterations
    float wreg[13];
    #pragma unroll
    for (int it = 0; it < 13; ++it) wreg[it] = W1pad[it * 256 + tid];

    #pragma unroll
    for (int it = 0; it < 3; ++it) {
        int e  = (it * 256 + tid) * 2;
        int rr = e / 50, cc = e - rr * 50;     // cc even -> 8B aligned in LDS
        *(float2*)&tile[rr * 52 + cc] = areg[it];
    }
    if (tid < 32) {
        int e  = (768 + tid) * 2;
        int rr = e / 50, cc = e - rr * 50;
        *(float2*)&tile[rr * 52 + cc] = atail;
        tile[tid * 52 + 50] = 0.f;
        tile[tid * 52 + 51] = 0.f;
    }
    #pragma unroll
    for (int it = 0; it < 13; ++it) wlds[it * 256 + tid] = wreg[it];
    __syncthreads();

    int mt = wave >> 2, nt = wave & 3;
    int ncl   = nt * 16 + (lane & 15);         // 0..63 (LDS col)
    int arow  = lane & 15;
    int khalf = 2 * (lane >> 4);
    const float* at = &tile[mt * 16 * 52];

    // ---- preload all 13 B fragments into registers (one-time) ----
    v2f bfrag[13];
    #pragma unroll
    for (int t = 0; t < 13; ++t) {
        int k = 4 * t + khalf;
        bfrag[t].x = wlds[k * 64 + ncl];
        bfrag[t].y = wlds[(k + 1) * 64 + ncl];
    }

    // ---- 13 back-to-back WMMAs, A fragments from LDS ----
    v8f acc = {};
    #pragma unroll
    for (int t = 0; t < 13; ++t) {
        int k = 4 * t + khalf;
        v2f a;
        a.x = at[arow * 52 + k];
        a.y = at[arow * 52 + k + 1];
        acc = __builtin_amdgcn_wmma_f32_16x16x4_f32(false, a, false, bfrag[t],
                                                    (short)0, acc, false, false);
    }

    // ---- epilogue: + Pj[b,j] + Pi[b,i] + rb, masked store to N=50 ----
    int ncolC = ncl < DEE ? ncl : (DEE - 1);   // clamped for safe loads
    bool colok = (ncl < DEE);
    float bv = rb[ncolC];
    #pragma unroll
    for (int r = 0; r < 8; ++r) {
        long m  = rowbase + mt * 16 + r + 8 * (lane >> 4); // flat (b*128+i)*128+j
        int  j  = (int)(m & 127);
        int  bi = (int)(m >> 7);                            // b*128 + i
        int  b_ = bi >> 7;
        int  i_ = bi & 127;
        float pj = P[((long)(b_ * SS + j )) * 112 + ncolC];        // Pj[b,j,c]
        float pi = P[((long)(b_ * SS + i_)) * 112 + 50 + ncolC];   // Pi[b,i,c]
        float val = acc[r] + pj + pi + bv;
        if (colok)
            eout[m * DEE + ncl] = val;
    }
}

// ---------------------------------------------------------------------------
extern "C" void kernel_launch(void* const* d_in, const int* in_sizes, int n_in,
                              void* d_out, int out_size, void* d_ws, size_t ws_size,
                              hipStream_t stream) {
    const float* wps  = (const float*)d_in[0];  // [8,128,128,50]
    const float* wadj = (const float*)d_in[1];  // [8,128,128,50]
    const float* x    = (const float*)d_in[2];  // [8,128,300]
    // d_in[3] self_loop is broadcast identity -> folded into k_reduce_wps
    const float* W_w  = (const float*)d_in[4];  // [300,300]
    const float* W_b  = (const float*)d_in[5];  // [300]
    const float* ln_a = (const float*)d_in[6];  // [300]
    const float* ln_b = (const float*)d_in[7];  // [300]
    const float* rW   = (const float*)d_in[8];  // [750,50]
    const float* rb   = (const float*)d_in[9];  // [50]

    float* out  = (float*)d_out;
    float* node = out;                                   // [8,128,300]
    float* eout = out + (size_t)BB * SS * DD;            // [8,128,128,50]

    float* ws    = (float*)d_ws;                // all offsets multiples of 4 floats
    float* Amat  = ws;                          // 131072
    float* xpad  = Amat  + 131072;              // 1024*320 = 327680
    float* Wwpad = xpad  + 327680;              // 320*320  = 102400
    float* Wbpad = Wwpad + 102400;              // 320
    float* C1    = Wbpad + 320;                 // 1024*320 = 327680
    float* C2    = C1    + 327680;              // 1024*320 = 327680
    float* X     = C2    + 327680;              // 1024*352 = 360448
    float* Wcat  = X     + 360448;              // 352*112  = 39424
    float* P     = Wcat  + 39424;               // 1024*112 = 114688
    float* W1pad = P     + 114688;              // 52*64    = 3328   (~6.9 MB total)

    // 0) one-shot pads (independent)
    k_pad_x_in<<<BB * SS, 256, 0, stream>>>(x, xpad);
    k_pad_w<<<(DP * DP + 255) / 256, 256, 0, stream>>>(W_w, W_b, Wwpad, Wbpad);
    k_pack_w1<<<13, 256, 0, stream>>>(rW, W1pad);
    k_pack_wcat<<<(352 * 112 + 255) / 256, 256, 0, stream>>>(rW, Wcat);

    // 1) A = mean_e(softmax) + I
    k_reduce_wps<<<BB * SS, 256, 0, stream>>>(wps, Amat);

    // 2) C1[b] = A[b](128x128) @ xpad[b](128x320)   — batched WMMA GEMM
    {
        dim3 g(SS / 64, DP / 16, BB);
        k_gemm_wmma<<<g, 128, 0, stream>>>(Amat, SS, (long)SS * SS,
                                           xpad, DP, (long)SS * DP,
                                           C1, DP, (long)SS * DP,
                                           SS, DP, SS, nullptr);
    }
    // 3) C2 = C1(1024x320) @ Wwpad(320x320) + Wbpad   (pad cols are exact zeros)
    {
        dim3 g((BB * SS) / 64, DP / 16, 1);
        k_gemm_wmma<<<g, 128, 0, stream>>>(C1, DP, 0, Wwpad, DP, 0, C2, DP, 0,
                                           BB * SS, DP, DP, Wbpad);
    }
    // 4) node = relu(layernorm(C2)) -> first output
    k_layernorm_relu<<<BB * SS, 128, 0, stream>>>(C2, ln_a, ln_b, node);

    // 5) X = [diag | node | pad]
    k_pack_x<<<BB * SS, 128, 0, stream>>>(wadj, node, X);

    // 6) P = X(1024x352) @ Wcat(352x112)   -> Pj | Pi
    {
        dim3 g((BB * SS) / 64, 112 / 16, 1);
        k_gemm_wmma<<<g, 128, 0, stream>>>(X, 352, 0, Wcat, 112, 0, P, 112, 0,
                                           BB * SS, 112, 352, nullptr);
    }
    // 7) edge_out = wadj @ W1 + Pj[b,j] + Pi[b,i] + rb   (dominant kernel)
    k_edge_out<<<(BB * SS * SS) / 32, 256, 0, stream>>>(wadj, W1pad, P, rb, eout);
}